// Analyzer_55241869361650
// MI455X (gfx1250) — compile-verified
//
#include <hip/hip_runtime.h>
#include <hip/hip_bf16.h>

typedef float v2f __attribute__((ext_vector_type(2)));
typedef float v8f __attribute__((ext_vector_type(8)));

// ---------------- problem constants ----------------
#define NPMAX 8192           // 32*32*8 cells per (b,e) group
#define NGROUP 16            // 8 batches * 2 elements
#define NTILE 512            // NPMAX/16

#define LATX 0.78125f        // 25/32
#define LATY 0.78125f
#define LATZ 0.375f          // 3/8
#define PADV 1e30f

// ---------------- workspace layout (units: 4-byte words) ----------------
static constexpr int O_PREDC = 0;
static constexpr int O_TGTC  = O_PREDC + NPMAX * 4;
static constexpr int O_PPOS  = O_TGTC  + NPMAX * 4;
static constexpr int O_TPOS  = O_PPOS  + NPMAX * 4;
static constexpr int O_CONF  = O_TPOS  + NPMAX * 4;
static constexpr int O_KEYS  = O_CONF  + NPMAX;       // u64 keys -> 2*NPMAX words
static constexpr int O_SIDX  = O_KEYS  + NPMAX * 2;
static constexpr int O_SP    = O_SIDX  + NPMAX;
static constexpr int O_AAP   = O_SP    + NPMAX * 4;
static constexpr int O_AAS   = O_AAP   + NPMAX;
static constexpr int O_AAT   = O_AAS   + NPMAX;
static constexpr int O_SPSEL = O_AAT   + NPMAX;
static constexpr int O_AASEL = O_SPSEL + NPMAX * 4;
static constexpr int O_RESTR = O_AASEL + NPMAX;
static constexpr int O_CORR  = O_RESTR + NPMAX;
static constexpr int O_SEL   = O_CORR  + NPMAX;
static constexpr int O_RHIT  = O_SEL   + NPMAX;
static constexpr int O_RARG  = O_RHIT  + NPMAX;
static constexpr int O_RHITN = O_RARG  + NPMAX;
static constexpr int O_RARGN = O_RHITN + NPMAX;
static constexpr int O_TIA   = O_RARGN + NPMAX;
static constexpr int O_PIA   = O_TIA   + NPMAX;
static constexpr int O_TIN   = O_PIA   + NPMAX;
static constexpr int O_PIN   = O_TIN   + NPMAX;
static constexpr int O_CNT   = O_PIN   + NPMAX;       // [0]=Np [1]=Nt [2]=Nsel [3]=NtiA [4]=NtiN
static constexpr int GSTRIDE = O_CNT + 32;            // per-group stride (even -> u64 aligned)
static constexpr int O_SEGOFF = GSTRIDE * NGROUP;     // 160 segment offsets

__device__ __forceinline__ float* wsf(void* ws, int g, int off) {
    return (float*)ws + (size_t)g * GSTRIDE + off;
}
__device__ __forceinline__ int* wsi(void* ws, int g, int off) {
    return (int*)((float*)ws + (size_t)g * GSTRIDE + off);
}
__device__ __forceinline__ float groupDiam2(int g) {
    float d = ((g & 1) == 0) ? (0.74f * 1.4f) : (0.528f * 1.4f);
    return d * d;
}

// exclusive block scan over 256 threads
__device__ __forceinline__ int exscan256(int v, int* lds, int& total) {
    int tid = threadIdx.x;
    lds[tid] = v; __syncthreads();
    for (int o = 1; o < 256; o <<= 1) {
        int t = (tid >= o) ? lds[tid - o] : 0;
        __syncthreads();
        lds[tid] += t;
        __syncthreads();
    }
    total = lds[255];
    int incl = lds[tid];
    __syncthreads();
    return incl - v;
}

// ================= K1: mask compaction, positions, norms =================
__global__ void k_compact(const float* __restrict__ pred,
                          const float* __restrict__ tgt, void* ws) {
    __shared__ int lds[256];
    int g = blockIdx.x, tid = threadIdx.x;
    int b = g >> 1, e = g & 1;

    float* predc = wsf(ws, g, O_PREDC);
    float* tgtc  = wsf(ws, g, O_TGTC);
    float* ppos  = wsf(ws, g, O_PPOS);
    float* tpos  = wsf(ws, g, O_TPOS);
    float* conf  = wsf(ws, g, O_CONF);
    float* aaP   = wsf(ws, g, O_AAP);
    float* aaT   = wsf(ws, g, O_AAT);
    int*   cnt   = wsi(ws, g, O_CNT);

    // ---- predictions ----
    int c0 = tid * 32, n = 0;
    for (int cc = 0; cc < 32; ++cc) {
        size_t idx = (((size_t)(b * NPMAX + c0 + cc) * 2 + e) * 4);
        if (pred[idx + 3] > 0.5f) n++;
    }
    int total, off = exscan256(n, lds, total);
    int Np = total;
    int pos = off;
    for (int cc = 0; cc < 32; ++cc) {
        int c = c0 + cc;
        size_t idx = (((size_t)(b * NPMAX + c) * 2 + e) * 4);
        float p3 = pred[idx + 3];
        if (p3 > 0.5f) {
            float x = pred[idx], y = pred[idx + 1], z = pred[idx + 2];
            predc[pos * 4 + 0] = x; predc[pos * 4 + 1] = y;
            predc[pos * 4 + 2] = z; predc[pos * 4 + 3] = p3;
            conf[pos] = p3;
            int ci = c >> 8, cj = (c >> 3) & 31, ck = c & 7;
            ppos[pos * 4 + 0] = (x + (float)ci) * LATX;
            ppos[pos * 4 + 1] = (y + (float)cj) * LATY;
            ppos[pos * 4 + 2] = (z + (float)ck) * LATZ;
            ppos[pos * 4 + 3] = 0.f;
            pos++;
        }
    }
    if (tid == 0) cnt[0] = Np;
    for (int r = Np + tid; r < NPMAX; r += 256) {
        ppos[r * 4 + 0] = PADV; ppos[r * 4 + 1] = PADV;
        ppos[r * 4 + 2] = PADV; ppos[r * 4 + 3] = 0.f;
    }
    __syncthreads();
    for (int r = tid; r < NPMAX; r += 256) {
        float x = ppos[r * 4], y = ppos[r * 4 + 1], z = ppos[r * 4 + 2];
        aaP[r] = x * x + y * y + z * z;
    }
    __syncthreads();

    // ---- targets ----
    n = 0;
    for (int cc = 0; cc < 32; ++cc) {
        size_t idx = (((size_t)(b * NPMAX + c0 + cc) * 2 + e) * 4);
        if (tgt[idx + 3] > 0.5f) n++;
    }
    off = exscan256(n, lds, total);
    int Nt = total;
    pos = off;
    for (int cc = 0; cc < 32; ++cc) {
        int c = c0 + cc;
        size_t idx = (((size_t)(b * NPMAX + c) * 2 + e) * 4);
        float t3 = tgt[idx + 3];
        if (t3 > 0.5f) {
            float x = tgt[idx], y = tgt[idx + 1], z = tgt[idx + 2];
            tgtc[pos * 4 + 0] = x; tgtc[pos * 4 + 1] = y;
            tgtc[pos * 4 + 2] = z; tgtc[pos * 4 + 3] = t3;
            int ci = c >> 8, cj = (c >> 3) & 31, ck = c & 7;
            tpos[pos * 4 + 0] = (x + (float)ci) * LATX;
            tpos[pos * 4 + 1] = (y + (float)cj) * LATY;
            tpos[pos * 4 + 2] = (z + (float)ck) * LATZ;
            tpos[pos * 4 + 3] = 0.f;
            pos++;
        }
    }
    if (tid == 0) cnt[1] = Nt;
    for (int r = Nt + tid; r < NPMAX; r += 256) {
        tpos[r * 4 + 0] = PADV; tpos[r * 4 + 1] = PADV;
        tpos[r * 4 + 2] = PADV; tpos[r * 4 + 3] = 0.f;
    }
    __syncthreads();
    for (int r = tid; r < NPMAX; r += 256) {
        float x = tpos[r * 4], y = tpos[r * 4 + 1], z = tpos[r * 4 + 2];
        aaT[r] = x * x + y * y + z * z;
    }
}

// ================= K2: ascending confidence sort (bitonic, u64 keys) ======
__global__ void k_sort(void* ws) {
    int g = blockIdx.x, tid = threadIdx.x;
    float* conf = wsf(ws, g, O_CONF);
    float* ppos = wsf(ws, g, O_PPOS);
    float* sp   = wsf(ws, g, O_SP);
    float* aaS  = wsf(ws, g, O_AAS);
    int*   sidx = wsi(ws, g, O_SIDX);
    int    Np   = wsi(ws, g, O_CNT)[0];
    unsigned long long* keys =
        (unsigned long long*)((float*)ws + (size_t)g * GSTRIDE + O_KEYS);

    for (int r = tid; r < NPMAX; r += 256) {
        keys[r] = (r < Np)
            ? ((unsigned long long)__float_as_uint(conf[r]) << 32) | (unsigned)r
            : ~0ull;
    }
    __syncthreads();
    for (int k = 2; k <= NPMAX; k <<= 1) {
        for (int j = k >> 1; j > 0; j >>= 1) {
            for (int i = tid; i < NPMAX; i += 256) {
                int ixj = i ^ j;
                if (ixj > i) {
                    bool up = ((i & k) == 0);
                    unsigned long long a = keys[i], b2 = keys[ixj];
                    if ((a > b2) == up) { keys[i] = b2; keys[ixj] = a; }
                }
            }
            __syncthreads();
        }
    }
    for (int r = tid; r < NPMAX; r += 256) {
        if (r < Np) {
            int id = (int)(keys[r] & 0xffffffffu);
            sidx[r] = id;
            float x = ppos[id * 4], y = ppos[id * 4 + 1], z = ppos[id * 4 + 2];
            sp[r * 4 + 0] = x; sp[r * 4 + 1] = y; sp[r * 4 + 2] = z; sp[r * 4 + 3] = 0.f;
            aaS[r] = x * x + y * y + z * z;
        } else {
            sidx[r] = 0;
            sp[r * 4 + 0] = PADV; sp[r * 4 + 1] = PADV;
            sp[r * 4 + 2] = PADV; sp[r * 4 + 3] = 0.f;
            aaS[r] = PADV * PADV * 3.0f; // inf
        }
    }
}

// ============== K3/K4: NMS triangular counts via f32 WMMA tiles ===========
template <int WITH_W>
__device__ __forceinline__ void nms_count_pass(void* ws) {
    int wid = blockIdx.x;
    int g = wid >> 9, jt = wid & (NTILE - 1);
    int lane = threadIdx.x, n = lane & 15, half = lane >> 4;

    float* sp  = wsf(ws, g, O_SP);
    float* aaS = wsf(ws, g, O_AAS);
    int* restr = wsi(ws, g, O_RESTR);
    int* outv  = WITH_W ? wsi(ws, g, O_CORR) : restr;
    int  Np    = wsi(ws, g, O_CNT)[0];
    float d2t  = groupDiam2(g);

    int jg = jt * 16 + n;
    const float* pj = sp + (size_t)jg * 4;
    v2f bv; bv.x = half ? pj[2] : pj[0]; bv.y = half ? 0.f : pj[1];
    float aaj = aaS[jg];

    int cntc = 0;
    int nT = (Np + 15) >> 4;
    if (jt < nT) {
        for (int it = 0; it <= jt; ++it) {
            const float* pi = sp + (size_t)(it * 16 + n) * 4;
            if (it + 1 <= jt) __builtin_prefetch(sp + (size_t)((it + 1) * 16 + n) * 4, 0, 0);
            v2f av; av.x = half ? pi[2] : pi[0]; av.y = half ? 0.f : pi[1];
            v8f acc = {0.f, 0.f, 0.f, 0.f, 0.f, 0.f, 0.f, 0.f};
            acc = __builtin_amdgcn_wmma_f32_16x16x4_f32(
                false, av, false, bv, (short)0, acc, false, false);
#pragma unroll
            for (int r = 0; r < 8; ++r) {
                int ig = it * 16 + r + (half << 3);
                float d2 = aaS[ig] + aaj - 2.f * acc[r];
                bool hit = (ig < jg) && (d2 < d2t);
                if (WITH_W) hit = hit && (restr[ig] != 0);
                if (hit) cntc++;
            }
        }
    }
    cntc += __shfl_down(cntc, 16);
    if (half == 0) outv[jg] = cntc;
}

__global__ void k_restrain(void* ws) { nms_count_pass<0>(ws); }
__global__ void k_correct(void* ws)  { nms_count_pass<1>(ws); }

// ============== K5: sel = (restrain==correct), compact survivors ==========
__global__ void k_selcompact(void* ws) {
    __shared__ int lds[256];
    int g = blockIdx.x, tid = threadIdx.x;
    int* restr = wsi(ws, g, O_RESTR);
    int* corr  = wsi(ws, g, O_CORR);
    int* selL  = wsi(ws, g, O_SEL);
    int* cnt   = wsi(ws, g, O_CNT);
    float* sp    = wsf(ws, g, O_SP);
    float* aaS   = wsf(ws, g, O_AAS);
    float* spSel = wsf(ws, g, O_SPSEL);
    float* aaSel = wsf(ws, g, O_AASEL);
    int Np = cnt[0];

    int c0 = tid * 32, n = 0;
    for (int cc = 0; cc < 32; ++cc) {
        int r = c0 + cc;
        if (r < Np && restr[r] == corr[r]) n++;
    }
    int total, off = exscan256(n, lds, total);
    int pos = off;
    for (int cc = 0; cc < 32; ++cc) {
        int r = c0 + cc;
        if (r < Np && restr[r] == corr[r]) selL[pos++] = r;
    }
    if (tid == 0) cnt[2] = total;
    __syncthreads();
    int Nsel = total;
    for (int s = tid; s < NPMAX; s += 256) {
        if (s < Nsel) {
            int r = selL[s];
            spSel[s * 4 + 0] = sp[r * 4 + 0]; spSel[s * 4 + 1] = sp[r * 4 + 1];
            spSel[s * 4 + 2] = sp[r * 4 + 2]; spSel[s * 4 + 3] = 0.f;
            aaSel[s] = aaS[r];
        } else {
            spSel[s * 4 + 0] = PADV; spSel[s * 4 + 1] = PADV;
            spSel[s * 4 + 2] = PADV; spSel[s * 4 + 3] = 0.f;
            aaSel[s] = PADV * PADV * 3.0f;
        }
    }
}

// ============== K6: target/pred matching via WMMA (any-hit + argmin) ======
__global__ void k_match(void* ws, int useNms) {
    int wid = blockIdx.x;
    int g = wid >> 9, rt = wid & (NTILE - 1);
    int lane = threadIdx.x, n = lane & 15, half = lane >> 4;

    float* tpos = wsf(ws, g, O_TPOS);
    float* aaT  = wsf(ws, g, O_AAT);
    int*   cnt  = wsi(ws, g, O_CNT);
    float* cp   = useNms ? wsf(ws, g, O_SPSEL) : wsf(ws, g, O_PPOS);
    float* aaC  = useNms ? wsf(ws, g, O_AASEL) : wsf(ws, g, O_AAP);
    int*   hit  = useNms ? wsi(ws, g, O_RHITN) : wsi(ws, g, O_RHIT);
    int*   arg  = useNms ? wsi(ws, g, O_RARGN) : wsi(ws, g, O_RARG);
    int Nt    = cnt[1];
    int Ncols = useNms ? cnt[2] : cnt[0];
    float d2t = groupDiam2(g);

    int rgbase = rt * 16;
    const float* pr = tpos + (size_t)(rgbase + n) * 4;
    v2f av; av.x = half ? pr[2] : pr[0]; av.y = half ? 0.f : pr[1];

    float best[8]; int bidx[8]; int any[8]; float aar[8];
#pragma unroll
    for (int r = 0; r < 8; ++r) {
        best[r] = 3.4e38f; bidx[r] = 0; any[r] = 0;
        aar[r] = aaT[rgbase + r + (half << 3)];
    }

    int nCT = (Ncols + 15) >> 4;
    if (rgbase < Nt) {
        for (int ct = 0; ct < nCT; ++ct) {
            const float* pc = cp + (size_t)(ct * 16 + n) * 4;
            if (ct + 1 < nCT) __builtin_prefetch(cp + (size_t)((ct + 1) * 16 + n) * 4, 0, 0);
            v2f bv; bv.x = half ? pc[2] : pc[0]; bv.y = half ? 0.f : pc[1];
            v8f acc = {0.f, 0.f, 0.f, 0.f, 0.f, 0.f, 0.f, 0.f};
            acc = __builtin_amdgcn_wmma_f32_16x16x4_f32(
                false, av, false, bv, (short)0, acc, false, false);
            int jg = ct * 16 + n;
#pragma unroll
            for (int r = 0; r < 8; ++r) {
                float d2 = aar[r] + aaC[jg] - 2.f * acc[r];
                if (jg < Ncols) {
                    if (d2 < d2t) any[r] = 1;
                    if (d2 < best[r]) { best[r] = d2; bidx[r] = jg; }
                }
            }
        }
    }
    // min-with-index + any reduction within each 16-lane half (wave32)
#pragma unroll
    for (int r = 0; r < 8; ++r) {
        for (int m = 8; m >= 1; m >>= 1) {
            float ov = __shfl_xor(best[r], m);
            int   oi = __shfl_xor(bidx[r], m);
            int   oa = __shfl_xor(any[r], m);
            if (ov < best[r] || (ov == best[r] && oi < bidx[r])) { best[r] = ov; bidx[r] = oi; }
            any[r] |= oa;
        }
    }
    if (n == 0) {
#pragma unroll
        for (int r = 0; r < 8; ++r) {
            int row = rgbase + r + (half << 3);
            hit[row] = any[r];
            arg[row] = bidx[r];
        }
    }
}

// ============== K7: compact ti/pi lists ==================================
__global__ void k_ticompact(void* ws) {
    __shared__ int lds[256];
    int g = blockIdx.x, tid = threadIdx.x;
    int* cnt = wsi(ws, g, O_CNT);
    int Nt = cnt[1];
    int c0 = tid * 32;
    for (int variant = 0; variant < 2; ++variant) {
        int* hit = variant ? wsi(ws, g, O_RHITN) : wsi(ws, g, O_RHIT);
        int* arg = variant ? wsi(ws, g, O_RARGN) : wsi(ws, g, O_RARG);
        int* ti  = variant ? wsi(ws, g, O_TIN)   : wsi(ws, g, O_TIA);
        int* pi  = variant ? wsi(ws, g, O_PIN)   : wsi(ws, g, O_PIA);
        int n = 0;
        for (int cc = 0; cc < 32; ++cc) {
            int r = c0 + cc;
            if (r < Nt && hit[r]) n++;
        }
        int total, off = exscan256(n, lds, total);
        int pos = off;
        for (int cc = 0; cc < 32; ++cc) {
            int r = c0 + cc;
            if (r < Nt && hit[r]) { ti[pos] = r; pi[pos] = arg[r]; pos++; }
        }
        if (tid == 0) cnt[3 + variant] = total;
        __syncthreads();
    }
}

// ============== K8: ragged output offsets (160 segments) ==================
__global__ void k_offsets(void* ws) {
    if (blockIdx.x != 0 || threadIdx.x != 0) return;
    int* seg = (int*)((float*)ws + O_SEGOFF);
    int off = 0;
    for (int g = 0; g < NGROUP; ++g) {
        int* cnt = wsi(ws, g, O_CNT);
        int Np = cnt[0], Nt = cnt[1], Nsel = cnt[2], NtiA = cnt[3], NtiN = cnt[4];
        int sz[10] = {Np * 4, Nsel * 4, Nt * 4, NtiA, NtiA, NtiN, NtiN,
                      Np * 3, Nsel * 3, Nt * 3};
        for (int t = 0; t < 10; ++t) { seg[g * 10 + t] = off; off += sz[t]; }
    }
}

// ============== K9: scatter into d_out ====================================
__global__ void k_scatter(void* ws, float* __restrict__ out) {
    int g = blockIdx.x, tid = threadIdx.x;
    int* seg = (int*)((float*)ws + O_SEGOFF) + g * 10;
    int* cnt = wsi(ws, g, O_CNT);
    int Np = cnt[0], Nt = cnt[1], Nsel = cnt[2], NtiA = cnt[3], NtiN = cnt[4];
    float* predc = wsf(ws, g, O_PREDC);
    float* tgtc  = wsf(ws, g, O_TGTC);
    float* ppos  = wsf(ws, g, O_PPOS);
    float* tpos  = wsf(ws, g, O_TPOS);
    float* spSel = wsf(ws, g, O_SPSEL);
    int* sidx = wsi(ws, g, O_SIDX);
    int* selL = wsi(ws, g, O_SEL);
    int* tiA = wsi(ws, g, O_TIA), *piA = wsi(ws, g, O_PIA);
    int* tiN = wsi(ws, g, O_TIN), *piN = wsi(ws, g, O_PIN);

    for (int x = tid; x < Np * 4; x += 256) out[seg[0] + x] = predc[x];
    for (int s = tid; s < Nsel; s += 256) {
        int row = sidx[selL[s]];
        for (int c = 0; c < 4; ++c) out[seg[1] + s * 4 + c] = predc[row * 4 + c];
    }
    for (int x = tid; x < Nt * 4; x += 256) out[seg[2] + x] = tgtc[x];
    for (int s = tid; s < NtiA; s += 256) {
        out[seg[3] + s] = (float)tiA[s];
        out[seg[4] + s] = (float)piA[s];
    }
    for (int s = tid; s < NtiN; s += 256) {
        out[seg[5] + s] = (float)tiN[s];
        out[seg[6] + s] = (float)piN[s];
    }
    for (int x = tid; x < Np * 3; x += 256)
        out[seg[7] + x] = ppos[(x / 3) * 4 + (x % 3)];
    for (int x = tid; x < Nsel * 3; x += 256)
        out[seg[8] + x] = spSel[(x / 3) * 4 + (x % 3)];
    for (int x = tid; x < Nt * 3; x += 256)
        out[seg[9] + x] = tpos[(x / 3) * 4 + (x % 3)];
}

// ================= host launch =================
extern "C" void kernel_launch(void* const* d_in, const int* in_sizes, int n_in,
                              void* d_out, int out_size, void* d_ws, size_t ws_size,
                              hipStream_t stream) {
    (void)in_sizes; (void)n_in; (void)out_size;
    const size_t need = ((size_t)GSTRIDE * NGROUP + 256) * sizeof(float);
    if (ws_size < need) return;  // deterministic no-op if scratch insufficient

    const float* pred = (const float*)d_in[0];
    const float* tgt  = (const float*)d_in[1];
    float* out = (float*)d_out;

    k_compact<<<NGROUP, 256, 0, stream>>>(pred, tgt, d_ws);
    k_sort<<<NGROUP, 256, 0, stream>>>(d_ws);
    k_restrain<<<NGROUP * NTILE, 32, 0, stream>>>(d_ws);
    k_correct<<<NGROUP * NTILE, 32, 0, stream>>>(d_ws);
    k_selcompact<<<NGROUP, 256, 0, stream>>>(d_ws);
    k_match<<<NGROUP * NTILE, 32, 0, stream>>>(d_ws, 0);
    k_match<<<NGROUP * NTILE, 32, 0, stream>>>(d_ws, 1);
    k_ticompact<<<NGROUP, 256, 0, stream>>>(d_ws);
    k_offsets<<<1, 32, 0, stream>>>(d_ws);
    k_scatter<<<NGROUP, 256, 0, stream>>>(d_ws, out);
}